// JointBilateralFilter3d_24172075941982
// MI455X (gfx1250) — compile-verified
//
#include <hip/hip_runtime.h>
#include <cstdint>

// Joint bilateral filter 3D, volume [1,1,128,128,96], window 7x7x7 (R=3).
// Floor is the trans pipe: 1 quarter-rate v_exp_f32 per tap (539M taps).
// Each thread computes FOUR adjacent z-voxels: one 10-element z-strip of
// ds_load_b64 feeds 28 taps (voxel q uses v[j+q], spatial term shared 4-way),
// so VALU (VOPD dual-issued), DS and loop overhead all sit below the exp cost.
// Halo tile staged into LDS via CDNA5 async global->LDS loads (ASYNCcnt).

#define DIMX 128
#define DIMY 128
#define DIMZ 96
#define RAD  3
#define TBX  16                 // x-voxels per block
#define TBY  8                  // y-voxels per block
#define TBZ  8                  // z-voxels per block (4 per thread)
#define TZT  (TBZ / 4)          // z-threads per block = 2
#define VPT  4                  // z-voxels per thread
#define HX   (TBX + 2 * RAD)    // 22
#define HY   (TBY + 2 * RAD)    // 14
#define HZ   (TBZ + 2 * RAD)    // 14
#define TILE_N (HX * HY * HZ)   // 4312 float2 = 34.5 KB LDS
#define NTHREADS (TBX * TBY * TZT)  // 256 = 8 waves

__global__ __launch_bounds__(NTHREADS)
void jbf3d_kernel(const float* __restrict__ inp, const float* __restrict__ gui,
                  const float* __restrict__ psx, const float* __restrict__ psy,
                  const float* __restrict__ psz, const float* __restrict__ pcs,
                  float* __restrict__ out)
{
    // Interleaved tile: .x = guidance, .y = input  (one ds_load_b64 per element)
    __shared__ float2 tile[TILE_N];

    const int tz = threadIdx.x;          // 0..1 -> voxels z0 + 4*tz + {0,1,2,3}
    const int ly = threadIdx.y;
    const int lx = threadIdx.z;
    const int z0 = blockIdx.x * TBZ;
    const int y0 = blockIdx.y * TBY;
    const int x0 = blockIdx.z * TBX;
    const int tid = (lx * TBY + ly) * TZT + tz;

    // LDS byte address of tile[] (flat-pointer low 32 bits == LDS address)
    const unsigned lds_base = (unsigned)(uintptr_t)(&tile[0]);

    // ---- Stage halo tile with async global->LDS loads (edge clamp here) ----
    for (int idx = tid; idx < TILE_N; idx += NTHREADS) {
        int cz = idx % HZ;
        int t  = idx / HZ;
        int cy = t % HY;
        int cx = t / HY;
        int gx = min(max(x0 + cx - RAD, 0), DIMX - 1);
        int gy = min(max(y0 + cy - RAD, 0), DIMY - 1);
        int gz = min(max(z0 + cz - RAD, 0), DIMZ - 1);
        size_t goff = ((size_t)(gx * DIMY + gy)) * DIMZ + gz;
        unsigned la = lds_base + (unsigned)idx * 8u;
        const float* gp = gui + goff;
        const float* ip = inp + goff;
        asm volatile("global_load_async_to_lds_b32 %0, %1, off"
                     :: "v"(la), "v"(gp) : "memory");
        asm volatile("global_load_async_to_lds_b32 %0, %1, off"
                     :: "v"(la + 4u), "v"(ip) : "memory");
    }
    asm volatile("s_wait_asynccnt 0x0" ::: "memory");  // drain this wave's async ops
    __syncthreads();                                   // cross-wave visibility

    // ---- Per-axis spatial log2-space terms:  a[d] = d^2 * 0.5 * log2(e) / sigma^2 ----
    const float L2E = 1.4426950408889634f;
    const float s_x = psx[0], s_y = psy[0], s_z = psz[0], c_s = pcs[0];
    const float kx = 0.5f * L2E / (s_x * s_x);
    const float ky = 0.5f * L2E / (s_y * s_y);
    const float kz = 0.5f * L2E / (s_z * s_z);
    const float kc = 0.5f * L2E / (c_s * c_s);

    float ax[7], ay[7], az[7];
#pragma unroll
    for (int d = 0; d < 7; ++d) {
        float f = (float)(d - RAD);
        ax[d] = f * f * kx;
        ay[d] = f * f * ky;
        az[d] = f * f * kz;
    }

    const int cxc = lx + RAD, cyc = ly + RAD;
    const int czA = VPT * tz + RAD;                     // center of voxel 0
    const int ctr = (cxc * HY + cyc) * HZ + czA;
    float gc[VPT];
#pragma unroll
    for (int q = 0; q < VPT; ++q) gc[q] = tile[ctr + q].x;

    // ---- 343 taps x 4 voxels; 10-element z-strip serves all four ----
    float num[VPT] = {0.f, 0.f, 0.f, 0.f};
    float den[VPT] = {0.f, 0.f, 0.f, 0.f};
    for (int dx = 0; dx < 7; ++dx) {
        const float nsx = -ax[dx];
        const int bx = (cxc + dx - RAD) * HY;
        for (int dy = 0; dy < 7; ++dy) {
            const float nsxy = nsx - ay[dy];
            const int b = (bx + cyc + dy - RAD) * HZ + czA - RAD;
            float2 v[7 + VPT - 1];                      // 10
#pragma unroll
            for (int j = 0; j < 7 + VPT - 1; ++j) v[j] = tile[b + j];  // 10x ds_load_b64
#pragma unroll
            for (int j = 0; j < 7; ++j) {
                const float c = nsxy - az[j];           // spatial term shared 4-way
#pragma unroll
                for (int q = 0; q < VPT; ++q) {
                    float d = gc[q] - v[j + q].x;
                    float w = __builtin_amdgcn_exp2f(fmaf(d * d, -kc, c));
                    num[q] = fmaf(w, v[j + q].y, num[q]);
                    den[q] += w;
                }
            }
        }
    }

    const size_t o = ((size_t)((x0 + lx) * DIMY + (y0 + ly))) * DIMZ + z0 + VPT * tz;
    float4 r;
    r.x = num[0] / den[0];
    r.y = num[1] / den[1];
    r.z = num[2] / den[2];
    r.w = num[3] / den[3];
    *reinterpret_cast<float4*>(out + o) = r;            // global_store_b128
}

extern "C" void kernel_launch(void* const* d_in, const int* in_sizes, int n_in,
                              void* d_out, int out_size, void* d_ws, size_t ws_size,
                              hipStream_t stream) {
    (void)in_sizes; (void)n_in; (void)out_size; (void)d_ws; (void)ws_size;
    const float* inp = (const float*)d_in[0];
    const float* gui = (const float*)d_in[1];
    const float* sx  = (const float*)d_in[2];
    const float* sy  = (const float*)d_in[3];
    const float* sz  = (const float*)d_in[4];
    const float* cs  = (const float*)d_in[5];
    float* out = (float*)d_out;

    dim3 block(TZT, TBY, TBX);                       // 2x8x16 = 256 threads (8 waves)
    dim3 grid(DIMZ / TBZ, DIMY / TBY, DIMX / TBX);   // 12 x 16 x 8
    jbf3d_kernel<<<grid, block, 0, stream>>>(inp, gui, sx, sy, sz, cs, out);
}